// SolveSchedulingCvxpyLayer_69080253989323
// MI455X (gfx1250) — compile-verified
//
#include <hip/hip_runtime.h>

// FISTA dual solver for the ramp-constrained QP, one wave32 per batch item.
// N=128 horizon -> 4 contiguous elements per lane. 4000 iterations fully in
// registers; cross-lane shifts via ds_bpermute; inputs staged via CDNA5
// global_load_async_to_lds_b128. Soft-threshold via v_med3_f32:
//   soft(s,t) = s - med3(s,-t,+t); lane31 elem3 uses t=+inf so the pad
// element of nu stays exactly 0 with no per-iteration mask.

#define N_HORIZON 128
#define C_RAMP    0.5f
#define N_ITERS   4000
#define WAVES_PER_BLOCK 8
#define THREADS_PER_BLOCK (WAVES_PER_BLOCK * 32)

__device__ __forceinline__ float bperm_f(int byteIdx, float v) {
    return __int_as_float(__builtin_amdgcn_ds_bpermute(byteIdx, __float_as_int(v)));
}

// ---- CDNA5 async global->LDS staging -------------------------------------
typedef int v4i __attribute__((vector_size(16)));

__device__ __forceinline__ void async_copy_b128(const float* g, const float* l) {
#if __has_builtin(__builtin_amdgcn_global_load_async_to_lds_b128)
    typedef __attribute__((address_space(1))) v4i* gp_t;
    typedef __attribute__((address_space(3))) v4i* lp_t;
    __builtin_amdgcn_global_load_async_to_lds_b128(
        (gp_t)(unsigned long long)g,
        (lp_t)(unsigned int)(unsigned long long)l,
        0, 0);
#else
    unsigned int lo = (unsigned int)(unsigned long long)l;  // flat -> LDS offset
    asm volatile("global_load_async_to_lds_b128 %0, %1, off"
                 :: "v"(lo), "v"(g) : "memory");
#endif
}

__device__ __forceinline__ void wait_async0() {
#if __has_builtin(__builtin_amdgcn_s_wait_asynccnt)
    __builtin_amdgcn_s_wait_asynccnt(0);
    asm volatile("" ::: "memory");   // compiler barrier: keep LDS reads below
#else
    asm volatile("s_wait_asynccnt 0" ::: "memory");
#endif
}
// --------------------------------------------------------------------------

__global__ __launch_bounds__(THREADS_PER_BLOCK)
void fista_ramp_qp_kernel(const float* __restrict__ gz0,
                          const float* __restrict__ gmu,
                          const float* __restrict__ gdg,
                          const float* __restrict__ gd2g,
                          float* __restrict__ gout,
                          int nItems)
{
    __shared__ __align__(16) float stage[WAVES_PER_BLOCK * 4 * N_HORIZON]; // 16 KB
    const int lane = (int)(threadIdx.x & 31u);
    const int wid  = (int)(threadIdx.x >> 5);
    const int item = (int)blockIdx.x * WAVES_PER_BLOCK + wid;
    if (item >= nItems) return;   // wave-uniform exit

    float* sl = &stage[wid * (4 * N_HORIZON)];
    const int goff = item * N_HORIZON + lane * 4;
    const int loff = lane * 4;

    // Stage the four input arrays for this item into LDS (async, 16B/lane).
    async_copy_b128(gz0  + goff, sl + 0 * N_HORIZON + loff);
    async_copy_b128(gmu  + goff, sl + 1 * N_HORIZON + loff);
    async_copy_b128(gdg  + goff, sl + 2 * N_HORIZON + loff);
    async_copy_b128(gd2g + goff, sl + 3 * N_HORIZON + loff);
    wait_async0();

    const float4 vz0 = *(const float4*)(sl + 0 * N_HORIZON + loff);
    const float4 vmu = *(const float4*)(sl + 1 * N_HORIZON + loff);
    const float4 vdg = *(const float4*)(sl + 2 * N_HORIZON + loff);
    const float4 vd2 = *(const float4*)(sl + 3 * N_HORIZON + loff);

    // d = d2g + 1 ; p = dg - d2g*z0 - mu ; precompute -p/d and -1/d.
    float z0a[4] = {vz0.x, vz0.y, vz0.z, vz0.w};
    float mua[4] = {vmu.x, vmu.y, vmu.z, vmu.w};
    float dga[4] = {vdg.x, vdg.y, vdg.z, vdg.w};
    float d2a[4] = {vd2.x, vd2.y, vd2.z, vd2.w};

    float npr[4], nrinv[4];
    float dmin = 3.402823466e38f;
#pragma unroll
    for (int k = 0; k < 4; ++k) {
        float dd = d2a[k] + 1.0f;
        float ri = 1.0f / dd;                       // precise, one-time
        float p  = dga[k] - d2a[k] * z0a[k] - mua[k];
        npr[k]   = -p * ri;
        nrinv[k] = -ri;
        dmin = fminf(dmin, dd);
    }
    // wave-wide min(d) via xor-shuffle reduction
#pragma unroll
    for (int k = 16; k >= 1; k >>= 1)
        dmin = fminf(dmin, bperm_f((lane ^ k) << 2, dmin));
    const float alpha = dmin * 0.25f;         // min(d)/4
    const float thr   = alpha * C_RAMP;       // alpha * c_ramp
    const float nthr  = -thr;
    // Lane31 elem3 is the pad entry (index 127): infinite threshold ->
    // soft-threshold output identically 0, no per-iteration masking.
    const bool  lane31 = (lane == 31);
    const float thrHi  = lane31 ? __builtin_inff() : thr;
    const float nthrHi = -thrHi;

    // State: nu, y padded to 128 entries; index 127 (lane31 elem3) stays 0.
    float nu[4] = {0.f, 0.f, 0.f, 0.f};
    float y[4]  = {0.f, 0.f, 0.f, 0.f};
    float t = 1.0f;

    const int  idxUp = ((lane + 31) & 31) << 2;   // lane-1 (wraps; masked at 0)
    const int  idxDn = ((lane + 1) & 31) << 2;    // lane+1 (wraps; pad eats it)
    const bool lane0 = (lane == 0);

    for (int it = 0; it < N_ITERS; ++it) {
        // z = -(p + D^T y)/d   (D^T v)_i = v_i - v_{i-1}, padded v
        float yp = bperm_f(idxUp, y[3]);
        yp = lane0 ? 0.0f : yp;
        float z[4];
        z[0] = fmaf(y[0] - yp,   nrinv[0], npr[0]);
        z[1] = fmaf(y[1] - y[0], nrinv[1], npr[1]);
        z[2] = fmaf(y[2] - y[1], nrinv[2], npr[2]);
        z[3] = fmaf(y[3] - y[2], nrinv[3], npr[3]);
        float zn = bperm_f(idxDn, z[0]);          // lane31: absorbed by inf thr

        // s = y + alpha * D(z);  nu_new = soft(s, thr) = s - med3(s,-thr,thr)
        float s0 = fmaf(alpha, z[0] - z[1], y[0]);
        float s1 = fmaf(alpha, z[1] - z[2], y[1]);
        float s2 = fmaf(alpha, z[2] - z[3], y[2]);
        float s3 = fmaf(alpha, z[3] - zn,   y[3]);
        float n0 = s0 - __builtin_amdgcn_fmed3f(s0, nthr,   thr);
        float n1 = s1 - __builtin_amdgcn_fmed3f(s1, nthr,   thr);
        float n2 = s2 - __builtin_amdgcn_fmed3f(s2, nthr,   thr);
        float n3 = s3 - __builtin_amdgcn_fmed3f(s3, nthrHi, thrHi);

        // Nesterov momentum: t_next = sqrt(t^2 + 0.25) + 0.5  (== 0.5(1+sqrt(1+4t^2)))
        float tn = __builtin_amdgcn_sqrtf(fmaf(t, t, 0.25f)) + 0.5f;
        float c  = (t - 1.0f) * __builtin_amdgcn_rcpf(tn);
        y[0] = fmaf(c, n0 - nu[0], n0); nu[0] = n0;
        y[1] = fmaf(c, n1 - nu[1], n1); nu[1] = n1;
        y[2] = fmaf(c, n2 - nu[2], n2); nu[2] = n2;
        y[3] = fmaf(c, n3 - nu[3], n3); nu[3] = n3;
        t = tn;
    }

    // z* = -(p + D^T nu)/d
    float np = bperm_f(idxUp, nu[3]);
    np = lane0 ? 0.0f : np;
    float4 outv;
    outv.x = fmaf(nu[0] - np,    nrinv[0], npr[0]);
    outv.y = fmaf(nu[1] - nu[0], nrinv[1], npr[1]);
    outv.z = fmaf(nu[2] - nu[1], nrinv[2], npr[2]);
    outv.w = fmaf(nu[3] - nu[2], nrinv[3], npr[3]);
    *(float4*)(gout + goff) = outv;
}

extern "C" void kernel_launch(void* const* d_in, const int* in_sizes, int n_in,
                              void* d_out, int out_size, void* d_ws, size_t ws_size,
                              hipStream_t stream) {
    (void)n_in; (void)d_ws; (void)ws_size; (void)out_size;
    const float* z0  = (const float*)d_in[0];
    const float* mu  = (const float*)d_in[1];
    const float* dg  = (const float*)d_in[2];
    const float* d2g = (const float*)d_in[3];
    float* out = (float*)d_out;

    const int B = in_sizes[0] / N_HORIZON;                 // 4096
    const int grid = (B + WAVES_PER_BLOCK - 1) / WAVES_PER_BLOCK;
    hipLaunchKernelGGL(fista_ramp_qp_kernel, dim3(grid), dim3(THREADS_PER_BLOCK),
                       0, stream, z0, mu, dg, d2g, out, B);
}